// EGTLayer_48747878809942
// MI455X (gfx1250) — compile-verified
//
#include <hip/hip_runtime.h>
#include <math.h>

// ---------------------------------------------------------------------------
// EGT layer forward, MI455X (gfx1250), wave32, f32 WMMA 16x16x4 throughout.
// B=8, N=512, D=512, ED=32, H=8, DH=64
// All LDS fragment reads are single aligned b64 loads (k-pair-interleaved
// B-tile layout) so the WMMA inner loops need no register repacking.
// ---------------------------------------------------------------------------

typedef float v2f __attribute__((ext_vector_type(2)));
typedef float v8f __attribute__((ext_vector_type(8)));

#define BPAD 130  // B-tile k-pair stride: 64 cols * 2 + 2 pad (8B aligned)
#define WPAD 66   // edge-weight k-pair stride: 32 cols * 2 + 2 pad

__device__ __forceinline__ v8f wmma_f32(v2f a, v2f b, v8f c) {
    return __builtin_amdgcn_wmma_f32_16x16x4_f32(false, a, false, b, (short)0, c,
                                                 false, false);
}

__device__ __forceinline__ float gelu_f(float x) {
    return 0.5f * x * (1.0f + erff(x * 0.70710678118654752f));
}

__device__ __forceinline__ float sigmoid_f(float x) {
    return 1.0f / (1.0f + expf(-x));
}

__device__ __forceinline__ v8f v8f_zero() {
    v8f z = {0.f, 0.f, 0.f, 0.f, 0.f, 0.f, 0.f, 0.f};
    return z;
}

// ---------------------------------------------------------------------------
// Row LayerNorm, 512 columns, one wave per row, 8 rows per 256-thread block.
// ---------------------------------------------------------------------------
__global__ __launch_bounds__(256) void ln_rows512(const float* __restrict__ in,
                                                  float* __restrict__ out,
                                                  const float* __restrict__ g,
                                                  const float* __restrict__ b) {
    const int row  = blockIdx.x * 8 + (threadIdx.x >> 5);
    const int lane = threadIdx.x & 31;
    const float* p = in + (size_t)row * 512;

    float4 f[4];
    float s = 0.f;
#pragma unroll
    for (int i = 0; i < 4; ++i) {
        f[i] = *(const float4*)(p + (i * 32 + lane) * 4);
        s += f[i].x + f[i].y + f[i].z + f[i].w;
    }
#pragma unroll
    for (int o = 16; o >= 1; o >>= 1) s += __shfl_xor(s, o, 32);
    const float m = s * (1.0f / 512.0f);

    float v = 0.f;
#pragma unroll
    for (int i = 0; i < 4; ++i) {
        float dx = f[i].x - m, dy = f[i].y - m, dz = f[i].z - m, dw = f[i].w - m;
        v += dx * dx + dy * dy + dz * dz + dw * dw;
    }
#pragma unroll
    for (int o = 16; o >= 1; o >>= 1) v += __shfl_xor(v, o, 32);
    const float rstd = rsqrtf(v * (1.0f / 512.0f) + 1e-5f);

    float* q = out + (size_t)row * 512;
#pragma unroll
    for (int i = 0; i < 4; ++i) {
        const int col = (i * 32 + lane) * 4;
        float4 o4;
        o4.x = (f[i].x - m) * rstd * g[col + 0] + b[col + 0];
        o4.y = (f[i].y - m) * rstd * g[col + 1] + b[col + 1];
        o4.z = (f[i].z - m) * rstd * g[col + 2] + b[col + 2];
        o4.w = (f[i].w - m) * rstd * g[col + 3] + b[col + 3];
        *(float4*)(q + col) = o4;
    }
}

// ---------------------------------------------------------------------------
// Generic tiled WMMA f32 GEMM: C = act(A(MxK) @ W(KxN) + bias) + resid
// Block tile 128x64 (8 waves x 16 rows), K-tile 32. All dims exact multiples.
// B tile staged k-pair-interleaved: (k,n) -> Bs[(k>>1)*BPAD + n*2 + (k&1)]
// ---------------------------------------------------------------------------
template <bool BIAS, bool RES, bool GELU>
__global__ __launch_bounds__(256) void gemm_nn(const float* __restrict__ A,
                                               const float* __restrict__ W,
                                               const float* __restrict__ bias,
                                               const float* __restrict__ resid,
                                               float* __restrict__ C, int Ncols,
                                               int K) {
    __shared__ alignas(16) float As[128 * 36];
    __shared__ alignas(16) float Bs[16 * BPAD];

    const int tid  = threadIdx.x;
    const int lane = tid & 31;
    const int wave = tid >> 5;
    const int tl   = lane & 15;
    const int th   = lane >> 4;
    const int m0   = blockIdx.y * 128;
    const int n0   = blockIdx.x * 64;

    v8f acc[4];
#pragma unroll
    for (int j = 0; j < 4; ++j) acc[j] = v8f_zero();

    const int kp = tid >> 4;          // k-pair 0..15
    const int n4 = (tid & 15) << 2;   // col group 0,4,..,60

    for (int k0 = 0; k0 < K; k0 += 32) {
#pragma unroll
        for (int i = 0; i < 4; ++i) {
            const int idx = tid + i * 256;
            const int r = idx >> 3, c4 = (idx & 7) << 2;
            float4 v = *(const float4*)(A + (size_t)(m0 + r) * K + k0 + c4);
            *(float4*)(&As[r * 36 + c4]) = v;
        }
        {
            float4 f0 = *(const float4*)(W + (size_t)(k0 + 2 * kp) * Ncols + n0 + n4);
            float4 f1 = *(const float4*)(W + (size_t)(k0 + 2 * kp + 1) * Ncols + n0 + n4);
            v2f p0 = {f0.x, f1.x}, p1 = {f0.y, f1.y};
            v2f p2 = {f0.z, f1.z}, p3 = {f0.w, f1.w};
            float* bp = &Bs[kp * BPAD + n4 * 2];
            *(v2f*)(bp + 0) = p0;
            *(v2f*)(bp + 2) = p1;
            *(v2f*)(bp + 4) = p2;
            *(v2f*)(bp + 6) = p3;
        }
        if (k0 + 32 < K) {
            __builtin_prefetch(A + (size_t)(m0 + (tid >> 1)) * K + k0 + 32, 0, 1);
            __builtin_prefetch(W + (size_t)(k0 + 32 + (tid >> 3)) * Ncols + n0, 0, 1);
        }
        __syncthreads();

#pragma unroll
        for (int ks = 0; ks < 8; ++ks) {
            const int kk = ks * 4 + 2 * th;
            v2f a = *(const v2f*)(&As[(wave * 16 + tl) * 36 + kk]);
#pragma unroll
            for (int j = 0; j < 4; ++j) {
                v2f bb = *(const v2f*)(&Bs[(kk >> 1) * BPAD + (j * 16 + tl) * 2]);
                acc[j] = wmma_f32(a, bb, acc[j]);
            }
        }
        __syncthreads();
    }

    const int row0 = m0 + wave * 16 + th * 8;
#pragma unroll
    for (int j = 0; j < 4; ++j) {
        const int gn = n0 + j * 16 + tl;
        const float bv = BIAS ? bias[gn] : 0.f;
#pragma unroll
        for (int r = 0; r < 8; ++r) {
            const int gm = row0 + r;
            float v = acc[j][r] + bv;
            if (GELU) v = gelu_f(v);
            if (RES) v += resid[(size_t)gm * Ncols + gn];
            C[(size_t)gm * Ncols + gn] = v;
        }
    }
}

// ---------------------------------------------------------------------------
// Edge LN2 + GE = edge_ln @ Wge + bge; E -> hhat buffer (seeds RMW), sigmoid(G)
// One thread per (b,l,m) position (row of 32).
// ---------------------------------------------------------------------------
__global__ __launch_bounds__(256) void edge_ln_ge(
    const float* __restrict__ edge_in, const float* __restrict__ g2,
    const float* __restrict__ b2, const float* __restrict__ Wge,
    const float* __restrict__ bge, float* __restrict__ hhat_e,
    float* __restrict__ gsig) {
    __shared__ float sW[32 * 16];
    __shared__ float sg[32], sb[32], sbg[16];
    const int tid = threadIdx.x;
    for (int i = tid; i < 512; i += 256) sW[i] = Wge[i];
    if (tid < 32) { sg[tid] = g2[tid]; sb[tid] = b2[tid]; }
    if (tid < 16) sbg[tid] = bge[tid];
    __syncthreads();

    const size_t p = (size_t)blockIdx.x * 256 + tid;
    const float* e = edge_in + p * 32;
    float x[32];
#pragma unroll
    for (int i = 0; i < 8; ++i) {
        float4 v = *(const float4*)(e + i * 4);
        x[i * 4 + 0] = v.x; x[i * 4 + 1] = v.y;
        x[i * 4 + 2] = v.z; x[i * 4 + 3] = v.w;
    }
    float s = 0.f;
#pragma unroll
    for (int k = 0; k < 32; ++k) s += x[k];
    const float m = s * (1.0f / 32.0f);
    float v2 = 0.f;
#pragma unroll
    for (int k = 0; k < 32; ++k) { const float d = x[k] - m; v2 += d * d; }
    const float rstd = rsqrtf(v2 * (1.0f / 32.0f) + 1e-5f);

    float acc[16];
#pragma unroll
    for (int n = 0; n < 16; ++n) acc[n] = sbg[n];
#pragma unroll
    for (int k = 0; k < 32; ++k) {
        const float xv = (x[k] - m) * rstd * sg[k] + sb[k];
#pragma unroll
        for (int n = 0; n < 16; ++n) acc[n] = fmaf(xv, sW[k * 16 + n], acc[n]);
    }
    float4 e0 = {acc[8], acc[9], acc[10], acc[11]};
    float4 e1 = {acc[12], acc[13], acc[14], acc[15]};
    *(float4*)(hhat_e + p * 8) = e0;
    *(float4*)(hhat_e + p * 8 + 4) = e1;
    float4 s0 = {sigmoid_f(acc[0]), sigmoid_f(acc[1]), sigmoid_f(acc[2]), sigmoid_f(acc[3])};
    float4 s1 = {sigmoid_f(acc[4]), sigmoid_f(acc[5]), sigmoid_f(acc[6]), sigmoid_f(acc[7])};
    *(float4*)(gsig + p * 8) = s0;
    *(float4*)(gsig + p * 8 + 4) = s1;
}

// ---------------------------------------------------------------------------
// Repack interleaved QKV (B,N,[3x(d*H+h)]) -> per-head (b,h,l,d) q/k/v buffers
// ---------------------------------------------------------------------------
__global__ __launch_bounds__(256) void repack_qkv(const float* __restrict__ qkv,
                                                  float* __restrict__ qb,
                                                  float* __restrict__ kb,
                                                  float* __restrict__ vb) {
    const size_t idx = (size_t)blockIdx.x * 256 + threadIdx.x;
    const int d = idx & 63;
    const int l = (idx >> 6) & 511;
    const int h = (idx >> 15) & 7;
    const int b = idx >> 18;
    const size_t src = ((size_t)(b * 512 + l)) * 1536 + d * 8 + h;
    qb[idx] = qkv[src];
    kb[idx] = qkv[src + 512];
    vb[idx] = qkv[src + 1024];
}

// ---------------------------------------------------------------------------
// Attention scores: per (b,h)  A_hat = Q_h(512x64) @ K_h^T, epilogue:
// hhat += clip(A_hat/8, -5, 5)   (hhat was pre-seeded with E)
// K tile staged transposed AND k-pair-interleaved.
// ---------------------------------------------------------------------------
__global__ __launch_bounds__(256) void attn_scores(const float* __restrict__ qbuf,
                                                   const float* __restrict__ kbuf,
                                                   float* __restrict__ hhat) {
    __shared__ alignas(16) float As[128 * 36];
    __shared__ alignas(16) float Bs[16 * BPAD];

    const int tid = threadIdx.x, lane = tid & 31, wave = tid >> 5;
    const int tl = lane & 15, th = lane >> 4;
    const int bh = blockIdx.z;
    const float* Q = qbuf + (size_t)bh * 512 * 64;
    const float* Kp = kbuf + (size_t)bh * 512 * 64;
    const int l0 = blockIdx.y * 128, mm0 = blockIdx.x * 64;

    v8f acc[4];
#pragma unroll
    for (int j = 0; j < 4; ++j) acc[j] = v8f_zero();

#pragma unroll
    for (int k0 = 0; k0 < 64; k0 += 32) {
#pragma unroll
        for (int i = 0; i < 4; ++i) {
            const int idx = tid + i * 256;
            const int r = idx >> 3, c4 = (idx & 7) << 2;
            float4 v = *(const float4*)(Q + (size_t)(l0 + r) * 64 + k0 + c4);
            *(float4*)(&As[r * 36 + c4]) = v;
        }
#pragma unroll
        for (int i = 0; i < 2; ++i) {
            const int idx = tid + i * 256;
            const int r = idx >> 3, c4 = (idx & 7) << 2;  // r = m-row 0..63
            float4 v = *(const float4*)(Kp + (size_t)(mm0 + r) * 64 + k0 + c4);
            v2f lo = {v.x, v.y}, hi = {v.z, v.w};
            *(v2f*)(&Bs[(c4 >> 1) * BPAD + r * 2]) = lo;
            *(v2f*)(&Bs[((c4 >> 1) + 1) * BPAD + r * 2]) = hi;
        }
        __syncthreads();
#pragma unroll
        for (int ks = 0; ks < 8; ++ks) {
            const int kk = ks * 4 + 2 * th;
            v2f a = *(const v2f*)(&As[(wave * 16 + tl) * 36 + kk]);
#pragma unroll
            for (int j = 0; j < 4; ++j) {
                v2f bb = *(const v2f*)(&Bs[(kk >> 1) * BPAD + (j * 16 + tl) * 2]);
                acc[j] = wmma_f32(a, bb, acc[j]);
            }
        }
        __syncthreads();
    }

    const int b = bh >> 3, h = bh & 7;
    const int row0 = l0 + wave * 16 + th * 8;
#pragma unroll
    for (int j = 0; j < 4; ++j) {
        const int gm = mm0 + j * 16 + tl;
#pragma unroll
        for (int r = 0; r < 8; ++r) {
            const int gl = row0 + r;
            float v = acc[j][r] * 0.125f;
            v = fminf(5.0f, fmaxf(-5.0f, v));
            const size_t pos = (((size_t)b * 512 + gl) * 512 + gm) * 8 + h;
            hhat[pos] = v + hhat[pos];  // += E (pre-seeded)
        }
    }
}

// ---------------------------------------------------------------------------
// Softmax over heads (axis of size 8) * sigmoid(G); write head-major (b,h,l,m)
// ---------------------------------------------------------------------------
__global__ __launch_bounds__(256) void softmax_gate(const float* __restrict__ hhat,
                                                    const float* __restrict__ gsig,
                                                    float* __restrict__ atild) {
    const size_t p = (size_t)blockIdx.x * 256 + threadIdx.x;
    float hv[8], gv[8];
    float4 a0 = *(const float4*)(hhat + p * 8);
    float4 a1 = *(const float4*)(hhat + p * 8 + 4);
    float4 g0 = *(const float4*)(gsig + p * 8);
    float4 g1 = *(const float4*)(gsig + p * 8 + 4);
    hv[0] = a0.x; hv[1] = a0.y; hv[2] = a0.z; hv[3] = a0.w;
    hv[4] = a1.x; hv[5] = a1.y; hv[6] = a1.z; hv[7] = a1.w;
    gv[0] = g0.x; gv[1] = g0.y; gv[2] = g0.z; gv[3] = g0.w;
    gv[4] = g1.x; gv[5] = g1.y; gv[6] = g1.z; gv[7] = g1.w;

    float mx = hv[0];
#pragma unroll
    for (int h = 1; h < 8; ++h) mx = fmaxf(mx, hv[h]);
    float s = 0.f;
#pragma unroll
    for (int h = 0; h < 8; ++h) { hv[h] = expf(hv[h] - mx); s += hv[h]; }
    const float inv = 1.0f / s;

    const int b = (int)(p >> 18);
    const int l = (int)(p >> 9) & 511;
    const int mc = (int)p & 511;
#pragma unroll
    for (int h = 0; h < 8; ++h) {
        atild[(((size_t)(b * 8 + h)) << 18) + ((size_t)l << 9) + mc] =
            hv[h] * inv * gv[h];
    }
}

// ---------------------------------------------------------------------------
// V_att: per (b,h)  A_tild_h(512x512) @ V_h(512x64) -> vatt (b,l,(d*8+h))
// ---------------------------------------------------------------------------
__global__ __launch_bounds__(256) void attn_apply(const float* __restrict__ atild,
                                                  const float* __restrict__ vbuf,
                                                  float* __restrict__ vatt) {
    __shared__ alignas(16) float As[128 * 36];
    __shared__ alignas(16) float Bs[16 * BPAD];

    const int tid = threadIdx.x, lane = tid & 31, wave = tid >> 5;
    const int tl = lane & 15, th = lane >> 4;
    const int bh = blockIdx.z;
    const float* A = atild + (size_t)bh * 512 * 512;
    const float* V = vbuf + (size_t)bh * 512 * 64;
    const int l0 = blockIdx.y * 128;

    v8f acc[4];
#pragma unroll
    for (int j = 0; j < 4; ++j) acc[j] = v8f_zero();

    const int kp = tid >> 4;
    const int n4 = (tid & 15) << 2;

    for (int k0 = 0; k0 < 512; k0 += 32) {
#pragma unroll
        for (int i = 0; i < 4; ++i) {
            const int idx = tid + i * 256;
            const int r = idx >> 3, c4 = (idx & 7) << 2;
            float4 v = *(const float4*)(A + (size_t)(l0 + r) * 512 + k0 + c4);
            *(float4*)(&As[r * 36 + c4]) = v;
        }
        {
            float4 f0 = *(const float4*)(V + (size_t)(k0 + 2 * kp) * 64 + n4);
            float4 f1 = *(const float4*)(V + (size_t)(k0 + 2 * kp + 1) * 64 + n4);
            v2f p0 = {f0.x, f1.x}, p1 = {f0.y, f1.y};
            v2f p2 = {f0.z, f1.z}, p3 = {f0.w, f1.w};
            float* bp = &Bs[kp * BPAD + n4 * 2];
            *(v2f*)(bp + 0) = p0;
            *(v2f*)(bp + 2) = p1;
            *(v2f*)(bp + 4) = p2;
            *(v2f*)(bp + 6) = p3;
        }
        __syncthreads();
#pragma unroll
        for (int ks = 0; ks < 8; ++ks) {
            const int kk = ks * 4 + 2 * th;
            v2f a = *(const v2f*)(&As[(wave * 16 + tl) * 36 + kk]);
#pragma unroll
            for (int j = 0; j < 4; ++j) {
                v2f bb = *(const v2f*)(&Bs[(kk >> 1) * BPAD + (j * 16 + tl) * 2]);
                acc[j] = wmma_f32(a, bb, acc[j]);
            }
        }
        __syncthreads();
    }

    const int b = bh >> 3, h = bh & 7;
    const int row0 = l0 + wave * 16 + th * 8;
#pragma unroll
    for (int j = 0; j < 4; ++j) {
        const int gd = j * 16 + tl;
#pragma unroll
        for (int r = 0; r < 8; ++r) {
            const int gl = row0 + r;
            vatt[((size_t)(b * 512 + gl)) * 512 + gd * 8 + h] = acc[j][r];
        }
    }
}

// ---------------------------------------------------------------------------
// Fused edge output: per 16-position chunk (one wave):
//   edge1 = H_hat(16x8) @ Woe(8x32) + boe + LN2(edge_in)      [WMMA]
//   x     = LN4(edge1)
//   y     = gelu(x @ We1 + be1)                               [WMMA]
//   out   = y @ We2 + be2 + edge1                             [WMMA]
// LN2 recomputed here so edge_ln never hits HBM.
// Weights staged pair-interleaved so B frags are single b64 LDS loads.
// ---------------------------------------------------------------------------
#define E2_WAVES 4
#define E2_CH 4
__global__ __launch_bounds__(128) void edge_out_fused(
    const float* __restrict__ edge_in, const float* __restrict__ hhat,
    const float* __restrict__ Woe, const float* __restrict__ boe,
    const float* __restrict__ ln2_g, const float* __restrict__ ln2_b,
    const float* __restrict__ ln4_g, const float* __restrict__ ln4_b,
    const float* __restrict__ We1, const float* __restrict__ be1,
    const float* __restrict__ We2, const float* __restrict__ be2,
    float* __restrict__ out_edge) {
    __shared__ alignas(16) float sWoe[4 * WPAD];
    __shared__ alignas(16) float sW1[16 * WPAD];
    __shared__ alignas(16) float sW2[16 * WPAD];
    __shared__ float sboe[32], sbe1[32], sbe2[32];
    __shared__ float sg2[32], sb2[32], sg4[32], sb4[32];
    __shared__ alignas(16) float hh[E2_WAVES][16 * 8];
    __shared__ alignas(16) float eln[E2_WAVES][16 * 36];
    __shared__ alignas(16) float xt[E2_WAVES][16 * 36];
    __shared__ alignas(16) float yt[E2_WAVES][16 * 36];

    const int tid = threadIdx.x, lane = tid & 31, w = tid >> 5;
    const int tl = lane & 15, th = lane >> 4;

    // pair-interleave weights: (k,n) -> [ (k>>1)*WPAD + n*2 + (k&1) ]
    if (tid < 128) {
        const int kpw = tid >> 5, n = tid & 31;  // 4 k-pairs for Woe
        sWoe[kpw * WPAD + n * 2 + 0] = Woe[(2 * kpw) * 32 + n];
        sWoe[kpw * WPAD + n * 2 + 1] = Woe[(2 * kpw + 1) * 32 + n];
    }
    for (int i = tid; i < 512; i += 128) {
        const int kpw = i >> 5, n = i & 31;      // 16 k-pairs
        sW1[kpw * WPAD + n * 2 + 0] = We1[(2 * kpw) * 32 + n];
        sW1[kpw * WPAD + n * 2 + 1] = We1[(2 * kpw + 1) * 32 + n];
        sW2[kpw * WPAD + n * 2 + 0] = We2[(2 * kpw) * 32 + n];
        sW2[kpw * WPAD + n * 2 + 1] = We2[(2 * kpw + 1) * 32 + n];
    }
    if (tid < 32) {
        sboe[tid] = boe[tid]; sbe1[tid] = be1[tid]; sbe2[tid] = be2[tid];
        sg2[tid] = ln2_g[tid]; sb2[tid] = ln2_b[tid];
        sg4[tid] = ln4_g[tid]; sb4[tid] = ln4_b[tid];
    }
    __syncthreads();

    for (int cc = 0; cc < E2_CH; ++cc) {
        const size_t chunk = ((size_t)blockIdx.x * E2_WAVES + w) * E2_CH + cc;
        const size_t p0 = chunk * 16;

        // stage H_hat (16x8) and raw edge rows (16x32)
        *(float4*)(&hh[w][lane * 4]) = *(const float4*)(hhat + p0 * 8 + lane * 4);
#pragma unroll
        for (int i = 0; i < 4; ++i) {
            const int idx = lane + i * 32;
            float4 v = *(const float4*)(edge_in + p0 * 32 + (size_t)idx * 4);
            const int r = idx >> 3, c = (idx & 7) << 2;
            *(float4*)(&eln[w][r * 36 + c]) = v;
        }
        __syncthreads();

        // LN2 per row (recompute; residual basis)
        if (lane < 16) {
            float* row = &eln[w][lane * 36];
            float s = 0.f;
#pragma unroll
            for (int c = 0; c < 32; ++c) s += row[c];
            const float m = s * (1.0f / 32.0f);
            float v2 = 0.f;
#pragma unroll
            for (int c = 0; c < 32; ++c) { const float d = row[c] - m; v2 += d * d; }
            const float rstd = rsqrtf(v2 * (1.0f / 32.0f) + 1e-5f);
#pragma unroll
            for (int c = 0; c < 32; ++c)
                row[c] = (row[c] - m) * rstd * sg2[c] + sb2[c];
        }
        __syncthreads();

        // WMMA1: hh @ Woe + boe + eln  ->  eln (edge1 residual)
        v8f a0 = v8f_zero(), a1 = v8f_zero();
#pragma unroll
        for (int ks = 0; ks < 2; ++ks) {
            const int kk = ks * 4 + 2 * th;
            v2f av = *(const v2f*)(&hh[w][tl * 8 + kk]);
            v2f b0 = *(const v2f*)(&sWoe[(kk >> 1) * WPAD + tl * 2]);
            v2f b1 = *(const v2f*)(&sWoe[(kk >> 1) * WPAD + (16 + tl) * 2]);
            a0 = wmma_f32(av, b0, a0);
            a1 = wmma_f32(av, b1, a1);
        }
#pragma unroll
        for (int r = 0; r < 8; ++r) {
            const int row = r + 8 * th;
            const float e0 = a0[r] + sboe[tl] + eln[w][row * 36 + tl];
            const float e1 = a1[r] + sboe[16 + tl] + eln[w][row * 36 + 16 + tl];
            eln[w][row * 36 + tl] = e0;
            eln[w][row * 36 + 16 + tl] = e1;
        }
        __syncthreads();

        // LN4 per row: eln -> xt
        if (lane < 16) {
            const float* row = &eln[w][lane * 36];
            float* xrow = &xt[w][lane * 36];
            float s = 0.f;
#pragma unroll
            for (int c = 0; c < 32; ++c) s += row[c];
            const float m = s * (1.0f / 32.0f);
            float v2 = 0.f;
#pragma unroll
            for (int c = 0; c < 32; ++c) { const float d = row[c] - m; v2 += d * d; }
            const float rstd = rsqrtf(v2 * (1.0f / 32.0f) + 1e-5f);
#pragma unroll
            for (int c = 0; c < 32; ++c)
                xrow[c] = (row[c] - m) * rstd * sg4[c] + sb4[c];
        }
        __syncthreads();

        // WMMA2: xt @ We1 + be1 -> gelu -> yt
        v8f c0 = v8f_zero(), c1 = v8f_zero();
#pragma unroll
        for (int ks = 0; ks < 8; ++ks) {
            const int kk = ks * 4 + 2 * th;
            v2f av = *(const v2f*)(&xt[w][tl * 36 + kk]);
            v2f b0 = *(const v2f*)(&sW1[(kk >> 1) * WPAD + tl * 2]);
            v2f b1 = *(const v2f*)(&sW1[(kk >> 1) * WPAD + (16 + tl) * 2]);
            c0 = wmma_f32(av, b0, c0);
            c1 = wmma_f32(av, b1, c1);
        }
#pragma unroll
        for (int r = 0; r < 8; ++r) {
            const int row = r + 8 * th;
            yt[w][row * 36 + tl] = gelu_f(c0[r] + sbe1[tl]);
            yt[w][row * 36 + 16 + tl] = gelu_f(c1[r] + sbe1[16 + tl]);
        }
        __syncthreads();

        // WMMA3: yt @ We2 + be2 + eln -> xt (final)
        v8f d0 = v8f_zero(), d1 = v8f_zero();
#pragma unroll
        for (int ks = 0; ks < 8; ++ks) {
            const int kk = ks * 4 + 2 * th;
            v2f av = *(const v2f*)(&yt[w][tl * 36 + kk]);
            v2f b0 = *(const v2f*)(&sW2[(kk >> 1) * WPAD + tl * 2]);
            v2f b1 = *(const v2f*)(&sW2[(kk >> 1) * WPAD + (16 + tl) * 2]);
            d0 = wmma_f32(av, b0, d0);
            d1 = wmma_f32(av, b1, d1);
        }
#pragma unroll
        for (int r = 0; r < 8; ++r) {
            const int row = r + 8 * th;
            xt[w][row * 36 + tl] = d0[r] + sbe2[tl] + eln[w][row * 36 + tl];
            xt[w][row * 36 + 16 + tl] = d1[r] + sbe2[16 + tl] + eln[w][row * 36 + 16 + tl];
        }
        __syncthreads();

        // coalesced store of the 16x32 output block
#pragma unroll
        for (int i = 0; i < 4; ++i) {
            const int idx = lane + i * 32;
            const int r = idx >> 3, c = (idx & 7) << 2;
            float4 v = *(const float4*)(&xt[w][r * 36 + c]);
            *(float4*)(out_edge + p0 * 32 + (size_t)idx * 4) = v;
        }
        __syncthreads();
    }
}

// ---------------------------------------------------------------------------
// Host orchestration
// ---------------------------------------------------------------------------
extern "C" void kernel_launch(void* const* d_in, const int* in_sizes, int n_in,
                              void* d_out, int out_size, void* d_ws,
                              size_t ws_size, hipStream_t stream) {
    const float* node = (const float*)d_in[0];
    const float* edge = (const float*)d_in[1];
    // d_in[2] = adj (unused)
    const float* ln1_g = (const float*)d_in[3];
    const float* ln1_b = (const float*)d_in[4];
    const float* ln2_g = (const float*)d_in[5];
    const float* ln2_b = (const float*)d_in[6];
    const float* Wqkv = (const float*)d_in[7];
    const float* bqkv = (const float*)d_in[8];
    const float* Wge = (const float*)d_in[9];
    const float* bge = (const float*)d_in[10];
    const float* Woh = (const float*)d_in[11];
    const float* boh = (const float*)d_in[12];
    const float* Woe = (const float*)d_in[13];
    const float* boe = (const float*)d_in[14];
    const float* ln3_g = (const float*)d_in[15];
    const float* ln3_b = (const float*)d_in[16];
    const float* Wn1 = (const float*)d_in[17];
    const float* bn1 = (const float*)d_in[18];
    const float* Wn2 = (const float*)d_in[19];
    const float* bn2 = (const float*)d_in[20];
    const float* ln4_g = (const float*)d_in[21];
    const float* ln4_b = (const float*)d_in[22];
    const float* We1 = (const float*)d_in[23];
    const float* be1 = (const float*)d_in[24];
    const float* We2 = (const float*)d_in[25];
    const float* be2 = (const float*)d_in[26];

    float* ws = (float*)d_ws;
    float* out_node = (float*)d_out;                 // (B,N,D)  2,097,152
    float* out_edge = (float*)d_out + 2097152;       // (B,N,N,ED)

    // workspace layout (floats)
    constexpr size_t O_NLN = 0;          // node_ln      2,097,152
    constexpr size_t O_QKV = 2097152;    // qkv          6,291,456
    constexpr size_t O_QB  = 8388608;    // qbuf         2,097,152
    constexpr size_t O_KB  = 10485760;   // kbuf         2,097,152
    constexpr size_t O_VB  = 12582912;   // vbuf         2,097,152
    constexpr size_t O_HH  = 14680064;   // hhat (E+clipQK)  16,777,216
    constexpr size_t O_GS  = 31457280;   // sigmoid(G)   16,777,216
    constexpr size_t O_AT  = 48234496;   // A_tild (b,h,l,m) 16,777,216
    constexpr size_t O_VA  = 65011712;   // V_att        2,097,152
    constexpr size_t O_N1  = 67108864;   // node after attn residual
    constexpr size_t O_L3  = 69206016;   // LN3(node)
    constexpr size_t O_FH  = 71303168;   // FFN hidden   4,194,304
    // total 75,497,472 floats = 302 MB

    // 1. node LN1
    ln_rows512<<<512, 256, 0, stream>>>(node, ws + O_NLN, ln1_g, ln1_b);
    // 2. QKV = node_ln @ Wqkv + bqkv   (4096 x 512 x 1536)
    gemm_nn<true, false, false><<<dim3(24, 32), 256, 0, stream>>>(
        ws + O_NLN, Wqkv, bqkv, nullptr, ws + O_QKV, 1536, 512);
    // 3. edge LN2 + GE; seeds hhat with E, gsig with sigmoid(G)
    edge_ln_ge<<<8192, 256, 0, stream>>>(edge, ln2_g, ln2_b, Wge, bge,
                                         ws + O_HH, ws + O_GS);
    // 4. per-head repack
    repack_qkv<<<8192, 256, 0, stream>>>(ws + O_QKV, ws + O_QB, ws + O_KB,
                                         ws + O_VB);
    // 5. hhat += clip(Q K^T / 8)
    attn_scores<<<dim3(8, 4, 64), 256, 0, stream>>>(ws + O_QB, ws + O_KB,
                                                    ws + O_HH);
    // 6. softmax over heads * sigmoid(G)
    softmax_gate<<<8192, 256, 0, stream>>>(ws + O_HH, ws + O_GS, ws + O_AT);
    // 7. V_att = A_tild @ V
    attn_apply<<<dim3(1, 4, 64), 256, 0, stream>>>(ws + O_AT, ws + O_VB,
                                                   ws + O_VA);
    // 8. node1 = V_att @ Woh + boh + node_ln
    gemm_nn<true, true, false><<<dim3(8, 32), 256, 0, stream>>>(
        ws + O_VA, Woh, boh, ws + O_NLN, ws + O_N1, 512, 512);
    // 9. LN3
    ln_rows512<<<512, 256, 0, stream>>>(ws + O_N1, ws + O_L3, ln3_g, ln3_b);
    // 10. hidden = gelu(LN3 @ Wn1 + bn1)
    gemm_nn<true, false, true><<<dim3(16, 32), 256, 0, stream>>>(
        ws + O_L3, Wn1, bn1, nullptr, ws + O_FH, 1024, 512);
    // 11. node_out = hidden @ Wn2 + bn2 + node1
    gemm_nn<true, true, false><<<dim3(8, 32), 256, 0, stream>>>(
        ws + O_FH, Wn2, bn2, ws + O_N1, out_node, 512, 1024);
    // 12. fused edge output (Woe + LN4 + FFN + residuals)
    edge_out_fused<<<8192, 128, 0, stream>>>(
        edge, ws + O_HH, Woe, boe, ln2_g, ln2_b, ln4_g, ln4_b, We1, be1, We2,
        be2, out_edge);
}